// Hypernetwork_PONG_hyper_55181739819744
// MI455X (gfx1250) — compile-verified
//
#include <hip/hip_runtime.h>
#include <hip/hip_bf16.h>
#include <math.h>

typedef __attribute__((ext_vector_type(2))) float v2f;
typedef __attribute__((ext_vector_type(8))) float v8f;

// ---------------------------------------------------------------------------
// Kernel 1: per needed pooled position (36 of them), compute
//   conv1(5x5x3x32,SAME)+relu -> maxpool(2, stride4)  [only 6x6 halo tile]
//   conv2(5x5x32x32,SAME)+relu -> maxpool(2, stride2) [only the 1 output]
// Result: A[pos][32] in workspace, padded to 48 rows (rows 36..47 = 0) so the
// WMMA GEMM can load unconditionally.
// ---------------------------------------------------------------------------
__global__ __launch_bounds__(256)
void conv_front_kernel(const float* __restrict__ x,
                       const float* __restrict__ k1,
                       const float* __restrict__ b1,
                       const float* __restrict__ k2,
                       const float* __restrict__ b2,
                       float* __restrict__ Aout) {
    __shared__ float sK1[2400];        // 5*5*3*32 conv1 weights
    __shared__ float sP1[6 * 6 * 32];  // conv1-pool tile (+SAME halo zeros)
    __shared__ float sC2[4 * 32];      // 2x2 conv2 outputs pre-pool

    const int tid = threadIdx.x;
    const int p   = blockIdx.x;

    if (p >= 36) {                     // zero-pad rows 36..47 of A
        if (tid < 32) Aout[p * 32 + tid] = 0.0f;
        return;
    }

    int bi, hc, wc;
    if (p < 32) {
        bi = p >> 2; hc = 20 * (p & 3); wc = 0;
    } else {
        const int tb[4] = {1, 2, 5, 6};
        const int th[4] = {26, 53, 26, 53};
        const int tw[4] = {53, 26, 53, 26};
        const int q = p - 32;
        bi = tb[q]; hc = th[q]; wc = tw[q];
    }

    for (int t = tid; t < 2400; t += 256) sK1[t] = k1[t];
    __syncthreads();

    // Phase A: 6x6 conv1-pool tile, 32 channels -> 1152 items
    for (int t = tid; t < 1152; t += 256) {
        const int co = t & 31;
        const int ij = t >> 5;
        const int i  = ij / 6;
        const int j  = ij - 6 * i;
        const int h1 = 2 * hc - 2 + i;   // 160-grid row
        const int w1 = 2 * wc - 2 + j;   // 160-grid col
        float v = 0.0f;                  // SAME zero-pad for conv2 input
        if (h1 >= 0 && h1 < 160 && w1 >= 0 && w1 < 160) {
            float mx = -3.4e38f;
            for (int a = 0; a < 2; ++a) {
                for (int bb = 0; bb < 2; ++bb) {
                    const int H1 = 4 * h1 + a;   // pool1 window [4h1,4h1+1]
                    const int W1 = 4 * w1 + bb;
                    float acc = b1[co];
                    for (int ky = 0; ky < 5; ++ky) {
                        const int Yp = H1 + ky - 2;
                        if (Yp < 0 || Yp >= 640) continue;
                        for (int kx = 0; kx < 5; ++kx) {
                            const int Xp = W1 + kx - 2;
                            if (Xp < 0 || Xp >= 640) continue;
                            const float* xp = x + (((size_t)bi * 640 + Yp) * 640 + Xp) * 3;
                            const float* kp = &sK1[(ky * 5 + kx) * 96 + co];
                            acc = fmaf(xp[0], kp[0],  acc);
                            acc = fmaf(xp[1], kp[32], acc);
                            acc = fmaf(xp[2], kp[64], acc);
                        }
                    }
                    mx = fmaxf(mx, fmaxf(acc, 0.0f));
                }
            }
            v = mx;
        }
        sP1[ij * 32 + co] = v;
    }
    __syncthreads();

    // Phase B: conv2 at the 2x2 pool window -> 128 items
    if (tid < 128) {
        const int co = tid & 31;
        const int ab = tid >> 5;        // 0..3
        const int a  = ab >> 1;
        const int bb = ab & 1;
        float acc = b2[co];
        for (int ky = 0; ky < 5; ++ky) {
            for (int kx = 0; kx < 5; ++kx) {
                const float* pp = &sP1[((a + ky) * 6 + (bb + kx)) * 32];
                const float* kk = k2 + (ky * 5 + kx) * 1024 + co;
                #pragma unroll 8
                for (int ci = 0; ci < 32; ++ci)
                    acc = fmaf(pp[ci], kk[ci * 32], acc);
            }
        }
        sC2[ab * 32 + co] = fmaxf(acc, 0.0f);
    }
    __syncthreads();

    if (tid < 32) {
        const float m0 = fmaxf(fmaxf(sC2[tid], sC2[32 + tid]),
                               fmaxf(sC2[64 + tid], sC2[96 + tid]));
        Aout[p * 32 + tid] = m0;
    }
}

// ---------------------------------------------------------------------------
// Kernel 2: d1 dense layer as WMMA f32 GEMM.
//   Y(36x300) = relu(A(48x32) @ W(32x304) + b), A pre-zero-padded.
// One wave32 per 16x16 tile; 8 x V_WMMA_F32_16X16X4_F32 K-steps.
// Branch-free inner loop: A loads are unconditional v2f (rows padded), B
// loads use a clamped address + value select (v_cndmask, no EXEC masking).
// ---------------------------------------------------------------------------
__global__ __launch_bounds__(32)
void d1_wmma_kernel(const float* __restrict__ A,     // 48x32, rows 36..47 = 0
                    const float* __restrict__ Wk,    // d1_k 32x300 row-major
                    const float* __restrict__ bias,  // 300
                    float* __restrict__ Yout) {      // 36x300
    const int tile = blockIdx.x;      // 0..56
    const int mt   = tile / 19;
    const int nt   = tile % 19;
    const int lane = threadIdx.x;     // 0..31
    const int half = lane >> 4;       // 0: K{0,1}; 1: K{2,3}
    const int l16  = lane & 15;

    const int m = mt * 16 + l16;              // A row for this lane (< 48)
    const int n = nt * 16 + l16;              // B col for this lane
    const bool nok = (n < 300);
    const int nc  = nok ? n : 299;            // clamped, always-legal address

    v8f c = {};
    #pragma unroll
    for (int k0 = 0; k0 < 8; ++k0) {
        const int kk = k0 * 4 + half * 2;
        const v2f a = *(const v2f*)(A + m * 32 + kk);   // aligned 8B (kk even)
        float b0 = Wk[kk * 300 + nc];
        float b1 = Wk[(kk + 1) * 300 + nc];
        v2f b;
        b.x = nok ? b0 : 0.0f;
        b.y = nok ? b1 : 0.0f;
        c = __builtin_amdgcn_wmma_f32_16x16x4_f32(
                /*neg_a=*/false, a, /*neg_b=*/false, b,
                /*c_mod=*/(short)0, c, /*reuse_a=*/false, /*reuse_b=*/false);
    }

    if (nok) {
        const float bv = bias[n];
        #pragma unroll
        for (int j = 0; j < 8; ++j) {
            const int mo = mt * 16 + half * 8 + j;   // C/D layout: VGPR j -> M=j / M=8+j
            if (mo < 36) {
                const float v = c[j] + bv;
                Yout[mo * 300 + n] = v > 0.0f ? v : 0.0f;
            }
        }
    }
}

// ---------------------------------------------------------------------------
// Kernel 3: the 102 needed d2 dot products. One wave per (pos, channel).
// ---------------------------------------------------------------------------
__global__ __launch_bounds__(32)
void d2_heads_kernel(const float* __restrict__ Y,    // 36x300
                     const float* __restrict__ Wk,   // d2_k 300x1340 row-major
                     const float* __restrict__ bias, // 1340
                     float* __restrict__ S) {        // 102 s-values
    const int job  = blockIdx.x;
    const int lane = threadIdx.x;

    int pos, ch;
    if (job < 32)      { pos = job;      ch = 1;   }
    else if (job < 64) { pos = job - 32; ch = 161; }
    else if (job < 96) { pos = job - 64; ch = 321; }
    else {
        const int p4[6] = {0, 32, 33, 16, 34, 35};
        const int c4[6] = {481, 491, 501, 481, 491, 501};
        pos = p4[job - 96]; ch = c4[job - 96];
    }

    const float* yrow = Y + pos * 300;
    float acc = 0.0f;
    for (int k = lane; k < 300; k += 32)
        acc = fmaf(yrow[k], Wk[k * 1340 + ch], acc);
    #pragma unroll
    for (int off = 16; off > 0; off >>= 1)
        acc += __shfl_down(acc, off, 32);
    if (lane == 0) S[job] = fmaxf(acc + bias[ch], 0.0f);
}

// ---------------------------------------------------------------------------
// Kernel 4: head expansion out[i] = relu(s_row * k[col] + b[col]).
//   o1 32x2001 | o2 32x801 | o3 32x321 | o4 6x65  -> 100326 total
// ---------------------------------------------------------------------------
__global__ __launch_bounds__(256)
void heads_expand_kernel(const float* __restrict__ S,
                         const float* __restrict__ w1k, const float* __restrict__ w1b,
                         const float* __restrict__ w2k, const float* __restrict__ w2b,
                         const float* __restrict__ w3k, const float* __restrict__ w3b,
                         const float* __restrict__ w4k, const float* __restrict__ w4b,
                         float* __restrict__ out) {
    const int i = blockIdx.x * blockDim.x + threadIdx.x;
    if (i >= 100326) return;
    float s, k, b;
    if (i < 64032) {
        const int r = i / 2001, c = i - r * 2001;
        s = S[r];       k = w1k[c]; b = w1b[c];
    } else if (i < 89664) {
        const int j = i - 64032;
        const int r = j / 801, c = j - r * 801;
        s = S[32 + r];  k = w2k[c]; b = w2b[c];
    } else if (i < 99936) {
        const int j = i - 89664;
        const int r = j / 321, c = j - r * 321;
        s = S[64 + r];  k = w3k[c]; b = w3b[c];
    } else {
        const int j = i - 99936;
        const int r = j / 65, c = j - r * 65;
        s = S[96 + r];  k = w4k[c]; b = w4b[c];
    }
    out[i] = fmaxf(fmaf(s, k, b), 0.0f);
}

// ---------------------------------------------------------------------------
extern "C" void kernel_launch(void* const* d_in, const int* in_sizes, int n_in,
                              void* d_out, int out_size, void* d_ws, size_t ws_size,
                              hipStream_t stream) {
    (void)in_sizes; (void)n_in; (void)out_size; (void)ws_size;

    const float* x   = (const float*)d_in[0];
    const float* k1  = (const float*)d_in[1];
    const float* b1  = (const float*)d_in[2];
    const float* k2  = (const float*)d_in[3];
    const float* b2  = (const float*)d_in[4];
    const float* d1k = (const float*)d_in[5];
    const float* d1b = (const float*)d_in[6];
    const float* d2k = (const float*)d_in[7];
    const float* d2b = (const float*)d_in[8];
    const float* w1k = (const float*)d_in[9];
    const float* w1b = (const float*)d_in[10];
    const float* w2k = (const float*)d_in[11];
    const float* w2b = (const float*)d_in[12];
    const float* w3k = (const float*)d_in[13];
    const float* w3b = (const float*)d_in[14];
    const float* w4k = (const float*)d_in[15];
    const float* w4b = (const float*)d_in[16];

    float* ws  = (float*)d_ws;
    float* A   = ws;            // 48*32   = 1536 floats (rows 36..47 zeroed)
    float* Yb  = ws + 1536;     // 36*300  = 10800 floats
    float* S   = ws + 12336;    // 102 floats
    float* out = (float*)d_out;

    conv_front_kernel<<<48, 256, 0, stream>>>(x, k1, b1, k2, b2, A);
    d1_wmma_kernel<<<57, 32, 0, stream>>>(A, d1k, d1b, Yb);
    d2_heads_kernel<<<102, 32, 0, stream>>>(Yb, d2k, d2b, S);
    heads_expand_kernel<<<(100326 + 255) / 256, 256, 0, stream>>>(
        S, w1k, w1b, w2k, w2b, w3k, w3b, w4k, w4b, out);
}